// AdditiveAttention_78494822301740
// MI455X (gfx1250) — compile-verified
//
#include <hip/hip_runtime.h>
#include <hip/hip_bf16.h>

typedef __attribute__((ext_vector_type(2))) float v2f;
typedef __attribute__((ext_vector_type(8))) float v8f;

#define BATCH 16
#define NQ    128
#define NKV   512
#define QS    256
#define KS    256
#define VS    256
#define HD    128

// tanh(x) = (e^{2x}-1)/(e^{2x}+1), via v_exp_f32 (exp2) + v_rcp_f32.
__device__ __forceinline__ float fast_tanh(float x) {
    x = fminf(fmaxf(x, -10.0f), 10.0f);
    float e = __builtin_amdgcn_exp2f(x * 2.8853900817779268f); // 2*log2(e)
    return (e - 1.0f) * __builtin_amdgcn_rcpf(e + 1.0f);
}

// Batched GEMM  C[b] (MxN) = A[b] (MxK) @ W (KxN), f32 row-major, dims compile-time.
// One wave32 computes a 16x64 output strip: 4 accumulators fed by ONE A fragment
// per k-step (4 v_wmma_f32_16x16x4_f32 per global_load_b64 of A).
// Block = 128 threads = 4 waves covering 4 consecutive M-tiles.
// grid = (N/64, M/64, B). All loads/stores use immediate offsets (K,N constant).
template <int M, int K, int N, bool W_BATCHED>
__global__ void wmma_gemm_f32(const float* __restrict__ A,
                              const float* __restrict__ W,
                              float* __restrict__ C) {
    const int lane = threadIdx.x & 31;
    const int wave = threadIdx.x >> 5;
    const int half = lane >> 4;   // 0: lanes 0-15, 1: lanes 16-31
    const int lq   = lane & 15;

    const int m0 = (blockIdx.y * 4 + wave) * 16;
    const int n0 = blockIdx.x * 64;
    const int b  = blockIdx.z;

    const float* __restrict__ Ab = A + (size_t)b * M * K;
    const float* __restrict__ Wb = W_BATCHED ? (W + (size_t)b * K * N) : W;
    float* __restrict__       Cb = C + (size_t)b * M * N;

    // A: lane -> row m0+(lane&15); VGPR pair holds K = k+2*half, k+2*half+1
    const float* __restrict__ Arow = Ab + (size_t)(m0 + lq) * K + 2 * half;
    // B: lane -> col n0+(lane&15)+16*nt; VGPR pair holds rows k+2*half, k+2*half+1
    const float* __restrict__ Wcol = Wb + (size_t)(2 * half) * N + (n0 + lq);

    v8f acc[4] = {};
    #pragma unroll 2
    for (int k = 0; k < K; k += 4) {
        v2f a = *(const v2f*)(Arow + k);           // global_load_b64, imm offset
        #pragma unroll
        for (int nt = 0; nt < 4; ++nt) {
            v2f bf;
            bf.x = Wcol[k * N + nt * 16];          // imm offsets (K,N constant)
            bf.y = Wcol[k * N + N + nt * 16];
            acc[nt] = __builtin_amdgcn_wmma_f32_16x16x4_f32(
                /*neg_a=*/false, a, /*neg_b=*/false, bf,
                /*c_mod=*/(short)0, acc[nt], /*reuse_a=*/false, /*reuse_b=*/false);
        }
    }

    // D layout: VGPR i -> row m0 + 8*half + i, col n0 + (lane&15) + 16*nt
    float* __restrict__ Cp = Cb + (size_t)(m0 + 8 * half) * N + (n0 + lq);
    #pragma unroll
    for (int nt = 0; nt < 4; ++nt) {
        #pragma unroll
        for (int i = 0; i < 8; ++i)
            Cp[i * N + nt * 16] = acc[nt][i];
    }
}

// scores[b,q,k] = sum_h wv[h]*tanh(qh[b,q,h] + kh[b,k,h]); mask; softmax -> attn.
// One 256-thread block per (q, b); each thread owns 2 of the 512 key positions.
__global__ void scores_softmax(const float* __restrict__ qh,
                               const float* __restrict__ kh,
                               const float* __restrict__ wv,
                               const int*   __restrict__ valid_lens,
                               float* __restrict__ attn) {
    __shared__ float q_s[HD];
    __shared__ float wv_s[HD];
    __shared__ float red[256];

    const int q = blockIdx.x;
    const int b = blockIdx.y;
    const int t = threadIdx.x;

    const float* __restrict__ qrow = qh + ((size_t)b * NQ + q) * HD;
    if (t < HD) {
        q_s[t]  = qrow[t];
        wv_s[t] = wv[t];
    }
    __syncthreads();

    const int vl = valid_lens[b];

    float s[2];
    #pragma unroll
    for (int j = 0; j < 2; ++j) {
        const int kidx = t + j * 256;
        const float4* __restrict__ krow =
            (const float4*)(kh + ((size_t)b * NKV + kidx) * HD);
        float acc = 0.0f;
        #pragma unroll 8
        for (int h4 = 0; h4 < HD / 4; ++h4) {
            float4 kv = krow[h4];
            const int h = h4 * 4;
            acc = fmaf(wv_s[h + 0], fast_tanh(q_s[h + 0] + kv.x), acc);
            acc = fmaf(wv_s[h + 1], fast_tanh(q_s[h + 1] + kv.y), acc);
            acc = fmaf(wv_s[h + 2], fast_tanh(q_s[h + 2] + kv.z), acc);
            acc = fmaf(wv_s[h + 3], fast_tanh(q_s[h + 3] + kv.w), acc);
        }
        s[j] = (kidx < vl) ? acc : -1000000.0f;
    }

    // block max
    red[t] = fmaxf(s[0], s[1]);
    __syncthreads();
    #pragma unroll
    for (int off = 128; off > 0; off >>= 1) {
        if (t < off) red[t] = fmaxf(red[t], red[t + off]);
        __syncthreads();
    }
    const float mx = red[0];
    __syncthreads();

    const float L2E = 1.4426950408889634f;
    const float e0 = __builtin_amdgcn_exp2f((s[0] - mx) * L2E);
    const float e1 = __builtin_amdgcn_exp2f((s[1] - mx) * L2E);

    // block sum
    red[t] = e0 + e1;
    __syncthreads();
    #pragma unroll
    for (int off = 128; off > 0; off >>= 1) {
        if (t < off) red[t] += red[t + off];
        __syncthreads();
    }
    const float inv = 1.0f / red[0];

    float* __restrict__ arow = attn + ((size_t)b * NQ + q) * NKV;
    arow[t]       = e0 * inv;
    arow[t + 256] = e1 * inv;
}

extern "C" void kernel_launch(void* const* d_in, const int* in_sizes, int n_in,
                              void* d_out, int out_size, void* d_ws, size_t ws_size,
                              hipStream_t stream) {
    const float* queries    = (const float*)d_in[0];  // (B, NQ, QS)
    const float* keys       = (const float*)d_in[1];  // (B, NKV, KS)
    const float* values     = (const float*)d_in[2];  // (B, NKV, VS)
    const int*   valid_lens = (const int*)  d_in[3];  // (B,)
    const float* Wq         = (const float*)d_in[4];  // (QS, H)
    const float* Wk         = (const float*)d_in[5];  // (KS, H)
    const float* wv         = (const float*)d_in[6];  // (H,)

    float* out  = (float*)d_out;                       // (B, NQ, VS)
    float* attn = out + (size_t)BATCH * NQ * VS;       // (B, NQ, NKV)

    float* qh = (float*)d_ws;                          // (B, NQ, H)
    float* kh = qh + (size_t)BATCH * NQ * HD;          // (B, NKV, H)

    // 1) qh = queries @ Wq   (shared W)
    wmma_gemm_f32<NQ, QS, HD, false>
        <<<dim3(HD / 64, NQ / 64, BATCH), 128, 0, stream>>>(queries, Wq, qh);

    // 2) kh = keys @ Wk      (shared W)
    wmma_gemm_f32<NKV, KS, HD, false>
        <<<dim3(HD / 64, NKV / 64, BATCH), 128, 0, stream>>>(keys, Wk, kh);

    // 3) scores + masked softmax -> attn (written into d_out's attn slot)
    scores_softmax<<<dim3(NQ, BATCH), 256, 0, stream>>>(qh, kh, wv, valid_lens, attn);

    // 4) out = attn @ values (per-batch W)
    wmma_gemm_f32<NQ, NKV, VS, true>
        <<<dim3(VS / 64, NQ / 64, BATCH), 128, 0, stream>>>(attn, values, out);
}